// GNN_78597901517024
// MI455X (gfx1250) — compile-verified
//
#include <hip/hip_runtime.h>
#include <hip/hip_bf16.h>

// ---------------------------------------------------------------------------
// GCN (4x GCNConv + linear head) for MI455X / gfx1250.
//   - Dense h@W via V_WMMA_F32_16X16X4_F32 (exact f32 matrix op, wave32).
//   - Edge gather/scale/scatter-add via wave-per-edge native f32 atomics
//     (global_atomic_add_f32; working set is L2-resident: 192 MB L2 >> 90 MB).
// ---------------------------------------------------------------------------

#define N_NODES 100000
#define N_EDGES 3200000

typedef float v2f __attribute__((ext_vector_type(2)));
typedef float v8f __attribute__((ext_vector_type(8)));

// Native f32 atomic add (global_atomic_add_f32) — avoids the CAS-loop fallback
// that plain atomicAdd(float*) takes without unsafe-fp-atomics.
__device__ __forceinline__ void fatomic_add(float* p, float v) {
#if defined(__HIP_PLATFORM_AMD__) || defined(__HIP__)
    unsafeAtomicAdd(p, v);
#else
    atomicAdd(p, v);
#endif
}

// ---------------- gcn_norm kernels ----------------

__global__ __launch_bounds__(256) void k_init_deg(float* __restrict__ deg, int n) {
    int i = blockIdx.x * blockDim.x + threadIdx.x;
    if (i < n) deg[i] = 1.0f;  // self-loop contributes 1 to every node's degree
}

__global__ __launch_bounds__(256) void k_count_deg(const long long* __restrict__ dst,
                                                   float* __restrict__ deg, int e) {
    int i = blockIdx.x * blockDim.x + threadIdx.x;
    if (i < e) fatomic_add(&deg[(int)dst[i]], 1.0f);
}

__global__ __launch_bounds__(256) void k_dinv(float* __restrict__ deg, int n) {
    int i = blockIdx.x * blockDim.x + threadIdx.x;
    if (i < n) {
        float d = deg[i];
        deg[i] = (d > 0.0f) ? rsqrtf(d) : 0.0f;   // in-place: deg -> dinv
    }
}

__global__ __launch_bounds__(256) void k_norm(const long long* __restrict__ src,
                                              const long long* __restrict__ dst,
                                              const float* __restrict__ dinv,
                                              float* __restrict__ norm, int e) {
    int i = blockIdx.x * blockDim.x + threadIdx.x;
    if (i < e) norm[i] = dinv[(int)src[i]] * dinv[(int)dst[i]];
}

// ---------------- dense GEMM: T[N,64] = A[N,KDIM] @ W[KDIM,64] ----------------
// One wave computes a 16x16 tile of T with V_WMMA_F32_16X16X4_F32.
// Block = 256 threads = 8 waves = 32 rows x 64 cols. Requires nrows % 32 == 0
// (N_NODES = 100000 = 3125 * 32).

template <int KDIM>
__global__ __launch_bounds__(256) void k_gemm_wmma(const float* __restrict__ A,
                                                   const float* __restrict__ W,
                                                   float* __restrict__ T) {
    const int lane = threadIdx.x & 31;
    const int wave = threadIdx.x >> 5;
    const int m0   = (blockIdx.x * 2 + (wave >> 2)) * 16;  // 16-row group
    const int n0   = (wave & 3) * 16;                      // 16-col group

    const int ml   = lane & 15;          // row within the 16-row group (A), col within group (B/D)
    const int koff = (lane >> 4) * 2;    // lanes 16..31 hold K+2,K+3 slices

    const int arow = m0 + ml;

    v8f acc = {};
#pragma unroll 4
    for (int k = 0; k < KDIM; k += 4) {
        // A fragment (16x4 f32): v0 = A[row][k+koff], v1 = A[row][k+koff+1]
        v2f a = *(const v2f*)(A + (long)arow * KDIM + k + koff);
        // B fragment (4x16 f32): v0 = W[k+koff][n0+ml], v1 = W[k+koff+1][n0+ml]
        v2f b;
        b.x = W[(k + koff + 0) * 64 + n0 + ml];
        b.y = W[(k + koff + 1) * 64 + n0 + ml];
        acc = __builtin_amdgcn_wmma_f32_16x16x4_f32(
            /*neg_a=*/false, a, /*neg_b=*/false, b,
            /*c_mod=*/(short)0, acc, /*reuse_a=*/false, /*reuse_b=*/false);
    }

    // D layout: VGPR r -> row m0+r (lanes 0-15) / m0+8+r (lanes 16-31), col n0+ml
    const int col   = n0 + ml;
    const int rbase = m0 + ((lane >> 4) << 3);
#pragma unroll
    for (int r = 0; r < 8; ++r) {
        T[(long)(rbase + r) * 64 + col] = acc[r];
    }
}

// ---------------- aggregation init: agg = t * dinv^2 (self-loop) + bias ----------------

__global__ __launch_bounds__(256) void k_agg_init(const float* __restrict__ t,
                                                  const float* __restrict__ dinv,
                                                  const float* __restrict__ bias,
                                                  float* __restrict__ agg, int total) {
    int i = blockIdx.x * blockDim.x + threadIdx.x;
    if (i < total) {
        int node = i >> 6;
        int c    = i & 63;
        float s  = dinv[node];
        agg[i] = t[i] * (s * s) + bias[c];
    }
}

// ---------------- edge scatter: agg[dst] += t[src] * norm ----------------
// One wave per edge; lane handles 2 consecutive floats of the 64-wide row.
// Gather is a wave-coalesced 256B read; scatter is native f32 atomics that
// the L2 coalesces per cacheline.

__global__ __launch_bounds__(256) void k_scatter(const long long* __restrict__ src,
                                                 const long long* __restrict__ dst,
                                                 const float* __restrict__ norm,
                                                 const float* __restrict__ t,
                                                 float* __restrict__ agg, int e) {
    int gid  = blockIdx.x * blockDim.x + threadIdx.x;
    int edge = gid >> 5;
    int lane = gid & 31;
    if (edge >= e) return;

    int s   = (int)src[edge];
    int d   = (int)dst[edge];
    float w = norm[edge];

    const float2 v = *(const float2*)(t + (long)s * 64 + lane * 2);
    float* p = agg + (long)d * 64 + lane * 2;
    fatomic_add(p + 0, v.x * w);
    fatomic_add(p + 1, v.y * w);
}

// ---------------- output head: out[i] = dot(y[i,:], Wout) + bout ----------------

__global__ __launch_bounds__(256) void k_out(const float* __restrict__ y,
                                             const float* __restrict__ Wout,
                                             const float* __restrict__ bout,
                                             float* __restrict__ out, int n) {
    int i = blockIdx.x * blockDim.x + threadIdx.x;
    if (i >= n) return;
    const float4* yr = (const float4*)(y + (long)i * 64);
    const float4* w4 = (const float4*)Wout;
    float acc = bout[0];
#pragma unroll
    for (int c = 0; c < 16; ++c) {
        float4 a = yr[c];
        float4 b = w4[c];
        acc += a.x * b.x + a.y * b.y + a.z * b.z + a.w * b.w;
    }
    out[i] = acc;
}

// ---------------- host-side orchestration ----------------

extern "C" void kernel_launch(void* const* d_in, const int* in_sizes, int n_in,
                              void* d_out, int out_size, void* d_ws, size_t ws_size,
                              hipStream_t stream) {
    (void)in_sizes; (void)n_in; (void)out_size; (void)ws_size;

    const float*     x    = (const float*)d_in[0];
    const long long* eidx = (const long long*)d_in[1];   // int64 [2, E] flat
    const long long* esrc = eidx;
    const long long* edst = eidx + N_EDGES;
    // d_in[2] = batch_index (unused)
    const float* W0   = (const float*)d_in[3];
    const float* b0   = (const float*)d_in[4];
    const float* W1   = (const float*)d_in[5];
    const float* b1   = (const float*)d_in[6];
    const float* W2   = (const float*)d_in[7];
    const float* b2   = (const float*)d_in[8];
    const float* W3   = (const float*)d_in[9];
    const float* b3   = (const float*)d_in[10];
    const float* Wout = (const float*)d_in[11];
    const float* bout = (const float*)d_in[12];

    float* out = (float*)d_out;           // [N, 1]
    float* y   = out + N_NODES;           // [N, 64] (second tuple output)

    float* ws   = (float*)d_ws;
    float* t    = ws;                          // [N, 64] GEMM output
    float* h    = ws + (long)N_NODES * 64;     // [N, 64] layer output / next input
    float* dinv = ws + (long)N_NODES * 128;    // [N] degree -> dinv (in place)
    float* norm = dinv + N_NODES;              // [E]

    const int TB = 256;
    const int gN    = (N_NODES + TB - 1) / TB;          // node-wise 1/thread
    const int gE    = (N_EDGES + TB - 1) / TB;          // edge-wise 1/thread
    const int gEw   = N_EDGES / 8;                      // edge-wise 1 wave/edge (E*32/256)
    const int gNF   = (N_NODES * 64) / TB;              // feature-wise
    const int gGEMM = N_NODES / 32;                     // 32 rows per block

    // --- normalization ---
    k_init_deg<<<gN, TB, 0, stream>>>(dinv, N_NODES);
    k_count_deg<<<gE, TB, 0, stream>>>(edst, dinv, N_EDGES);
    k_dinv<<<gN, TB, 0, stream>>>(dinv, N_NODES);
    k_norm<<<gE, TB, 0, stream>>>(esrc, edst, dinv, norm, N_EDGES);

    // --- layer 0: x[N,128] @ W0 -> t; agg -> h ---
    k_gemm_wmma<128><<<gGEMM, TB, 0, stream>>>(x, W0, t);
    k_agg_init<<<gNF, TB, 0, stream>>>(t, dinv, b0, h, N_NODES * 64);
    k_scatter<<<gEw, TB, 0, stream>>>(esrc, edst, norm, t, h, N_EDGES);

    // --- layer 1 ---
    k_gemm_wmma<64><<<gGEMM, TB, 0, stream>>>(h, W1, t);
    k_agg_init<<<gNF, TB, 0, stream>>>(t, dinv, b1, h, N_NODES * 64);
    k_scatter<<<gEw, TB, 0, stream>>>(esrc, edst, norm, t, h, N_EDGES);

    // --- layer 2 ---
    k_gemm_wmma<64><<<gGEMM, TB, 0, stream>>>(h, W2, t);
    k_agg_init<<<gNF, TB, 0, stream>>>(t, dinv, b2, h, N_NODES * 64);
    k_scatter<<<gEw, TB, 0, stream>>>(esrc, edst, norm, t, h, N_EDGES);

    // --- layer 3: aggregate directly into y = d_out + N ---
    k_gemm_wmma<64><<<gGEMM, TB, 0, stream>>>(h, W3, t);
    k_agg_init<<<gNF, TB, 0, stream>>>(t, dinv, b3, y, N_NODES * 64);
    k_scatter<<<gEw, TB, 0, stream>>>(esrc, edst, norm, t, y, N_EDGES);

    // --- head ---
    k_out<<<gN, TB, 0, stream>>>(y, Wout, bout, out, N_NODES);
}